// TO_GNN_61658550502081
// MI455X (gfx1250) — compile-verified
//
#include <hip/hip_runtime.h>
#include <hip/hip_bf16.h>

#define Nn   50000
#define Ee   800000
#define Gg   500
#define Ll   4
#define Hh   64
#define DINk 128
#define EPSv 1e-5f

typedef __attribute__((ext_vector_type(2))) float v2f;
typedef __attribute__((ext_vector_type(8))) float v8f;

enum { FUSE_NONE = 0, FUSE_BIAS = 1, FUSE_BIAS_RELU = 2 };

// ---------------------------------------------------------------------------
// WMMA f32 GEMM: Y[M,64] = X[M,K] @ W[K,64] (+ bias)(+relu)
// 256-thread block = 8 wave32s; each wave computes a 16x64 tile via 4
// accumulators of V_WMMA_F32_16X16X4_F32.
// W is staged once per block into LDS, transposed with +2 padding so each
// B fragment (two consecutive K values of one column) is one ds_load_b64.
// A layout (16x4 f32): lanes 0-15 row M=lane, VGPR0=K0,VGPR1=K1;
//                      lanes 16-31 same rows, VGPR0=K2,VGPR1=K3.
// B layout (4x16 f32): VGPR0: lanes0-15 K=0 / lanes16-31 K=2; VGPR1: K=1/K=3.
// C/D layout: VGPR r -> lanes0-15 M=r, lanes16-31 M=r+8; N=lane&15.
// ---------------------------------------------------------------------------
#define GEMM_WAVES 8
__global__ void gemm64_wmma(const float* __restrict__ X, const float* __restrict__ W,
                            const float* __restrict__ bias, float* __restrict__ Y,
                            int M, int K, int fuse) {
  __shared__ float Wt[Hh * (DINk + 2)];          // transposed W, padded stride
  const int KP = K + 2;

  // ---- cooperative staging: Wt[c][k] = W[k][c] ----
  for (int idx = threadIdx.x; idx < K * Hh; idx += blockDim.x) {
    const int k = idx >> 6, c = idx & 63;
    Wt[c * KP + k] = W[(size_t)k * Hh + c];
  }
  __syncthreads();

  const int waveId = threadIdx.x >> 5;
  const int lane   = threadIdx.x & 31;
  const int row0   = (blockIdx.x * GEMM_WAVES + waveId) << 4;
  if (row0 >= M) return;                         // wave-uniform, after barrier

  const int mrow  = lane & 15;
  const int khalf = (lane >> 4) << 1;            // 0 for lanes 0-15, 2 for 16-31

  const int   r     = row0 + mrow;
  const bool  rok   = (r < M);
  const int   rs    = rok ? r : (M - 1);         // clamp: load stays in-bounds
  const float amask = rok ? 1.0f : 0.0f;         // arithmetic masking, no EXEC churn
  const float* xrow = X + (size_t)rs * K;
  __builtin_prefetch(xrow, 0, 0);                // global_prefetch_b8

  const float* bt0 = &Wt[(      mrow) * KP + khalf];
  const float* bt1 = &Wt[(16 +  mrow) * KP + khalf];
  const float* bt2 = &Wt[(32 +  mrow) * KP + khalf];
  const float* bt3 = &Wt[(48 +  mrow) * KP + khalf];

  v8f acc0 = {}, acc1 = {}, acc2 = {}, acc3 = {};
  for (int k0 = 0; k0 < K; k0 += 4) {
    v2f a = *(const v2f*)(xrow + k0 + khalf);    // unconditional global_load_b64
    a[0] *= amask; a[1] *= amask;
    const v2f b0 = *(const v2f*)(bt0 + k0);      // ds_load_b64 each
    const v2f b1 = *(const v2f*)(bt1 + k0);
    const v2f b2 = *(const v2f*)(bt2 + k0);
    const v2f b3 = *(const v2f*)(bt3 + k0);
    acc0 = __builtin_amdgcn_wmma_f32_16x16x4_f32(false, a, false, b0, (short)0, acc0, false, false);
    acc1 = __builtin_amdgcn_wmma_f32_16x16x4_f32(false, a, false, b1, (short)0, acc1, false, false);
    acc2 = __builtin_amdgcn_wmma_f32_16x16x4_f32(false, a, false, b2, (short)0, acc2, false, false);
    acc3 = __builtin_amdgcn_wmma_f32_16x16x4_f32(false, a, false, b3, (short)0, acc3, false, false);
  }

  const int rowOff = (lane < 16) ? 0 : 8;
  for (int rr = 0; rr < 8; ++rr) {
    const int orow = row0 + rr + rowOff;
    if (orow < M) {
      float* yp = Y + (size_t)orow * Hh;
      float v0 = acc0[rr], v1 = acc1[rr], v2 = acc2[rr], v3 = acc3[rr];
      if (fuse >= FUSE_BIAS) {
        v0 += bias[mrow]; v1 += bias[16 + mrow]; v2 += bias[32 + mrow]; v3 += bias[48 + mrow];
      }
      if (fuse == FUSE_BIAS_RELU) {
        v0 = fmaxf(v0, 0.f); v1 = fmaxf(v1, 0.f); v2 = fmaxf(v2, 0.f); v3 = fmaxf(v3, 0.f);
      }
      yp[mrow] = v0; yp[16 + mrow] = v1; yp[32 + mrow] = v2; yp[48 + mrow] = v3;
    }
  }
}

// ---------------------------------------------------------------------------
// Elementwise / graph kernels
// ---------------------------------------------------------------------------
__global__ void fill_kernel(float* p, float v, int n) {
  int i = blockIdx.x * blockDim.x + threadIdx.x;
  if (i < n) p[i] = v;
}

__global__ void count_kernel(const int* __restrict__ idx, float* __restrict__ c, int E) {
  int e = blockIdx.x * blockDim.x + threadIdx.x;
  if (e < E) atomicAdd(&c[idx[e]], 1.0f);
}

__global__ void rsqrt_kernel(const float* __restrict__ deg, float* __restrict__ dinv, int n) {
  int i = blockIdx.x * blockDim.x + threadIdx.x;
  if (i < n) dinv[i] = rsqrtf(fmaxf(deg[i], 1.0f));
}

// acc[i][f] = bias[f] + dinv[i]^2 * h[i][f]   (self-loop term + gcn bias)
__global__ void init_acc_kernel(float* __restrict__ acc, const float* __restrict__ h,
                                const float* __restrict__ dinv, const float* __restrict__ bias,
                                int n) {
  int idx = blockIdx.x * blockDim.x + threadIdx.x;
  if (idx >= n * Hh) return;
  int i = idx >> 6, f = idx & 63;
  float di = dinv[i];
  acc[idx] = bias[f] + di * di * h[idx];
}

// acc[dst] += dinv[src]*dinv[dst]*h[src]  (16 threads/edge, float4 each)
__global__ void edge_agg_kernel(const float* __restrict__ h, const float* __restrict__ dinv,
                                const int* __restrict__ src, const int* __restrict__ dst,
                                float* __restrict__ acc, int E) {
  int tid = blockIdx.x * blockDim.x + threadIdx.x;
  if (tid >= E * 16) return;
  int e = tid >> 4;
  int f = (tid & 15) << 2;
  int s = src[e], d = dst[e];
  float nrm = dinv[s] * dinv[d];
  float4 hv = *(const float4*)(h + (size_t)s * Hh + f);
  float* ap = acc + (size_t)d * Hh + f;
  atomicAdd(ap + 0, hv.x * nrm);
  atomicAdd(ap + 1, hv.y * nrm);
  atomicAdd(ap + 2, hv.z * nrm);
  atomicAdd(ap + 3, hv.w * nrm);
}

__global__ void relu_kernel(const float* __restrict__ in, float* __restrict__ outp, int n) {
  int i = blockIdx.x * blockDim.x + threadIdx.x;
  if (i < n) outp[i] = fmaxf(in[i], 0.0f);
}

// ssum[src] += (Xc[src]-Xc[dst])^2 per feature
__global__ void edge_diff_kernel(const float* __restrict__ xc, const int* __restrict__ src,
                                 const int* __restrict__ dst, float* __restrict__ ssum, int E) {
  int tid = blockIdx.x * blockDim.x + threadIdx.x;
  if (tid >= E * 16) return;
  int e = tid >> 4;
  int f = (tid & 15) << 2;
  int s = src[e], d = dst[e];
  float4 a = *(const float4*)(xc + (size_t)s * Hh + f);
  float4 b = *(const float4*)(xc + (size_t)d * Hh + f);
  float* sp = ssum + (size_t)s * Hh + f;
  float dx = a.x - b.x, dy = a.y - b.y, dz = a.z - b.z, dw = a.w - b.w;
  atomicAdd(sp + 0, dx * dx);
  atomicAdd(sp + 1, dy * dy);
  atomicAdd(sp + 2, dz * dz);
  atomicAdd(sp + 3, dw * dw);
}

// tau = tanh(ssum/max(cnt,1)); x = (1-tau)*x + tau*xn
__global__ void combine_kernel(float* __restrict__ x, const float* __restrict__ xn,
                               const float* __restrict__ ssum, const float* __restrict__ cnt,
                               int n) {
  int idx = blockIdx.x * blockDim.x + threadIdx.x;
  if (idx >= n * Hh) return;
  int i = idx >> 6;
  float tau = tanhf(ssum[idx] / fmaxf(cnt[i], 1.0f));
  x[idx] = (1.0f - tau) * x[idx] + tau * xn[idx];
}

__global__ void pool_kernel(const float* __restrict__ x, const int* __restrict__ batch,
                            float* __restrict__ pooled, int n) {
  int tid = blockIdx.x * blockDim.x + threadIdx.x;
  if (tid >= n * 16) return;
  int i = tid >> 4;
  int f = (tid & 15) << 2;
  int g = batch[i];
  float4 v = *(const float4*)(x + (size_t)i * Hh + f);
  float* pp = pooled + (size_t)g * Hh + f;
  atomicAdd(pp + 0, v.x);
  atomicAdd(pp + 1, v.y);
  atomicAdd(pp + 2, v.z);
  atomicAdd(pp + 3, v.w);
}

// batch-stat BN: one thread per column over M rows
__global__ void bn_stats_kernel(const float* __restrict__ t, float* __restrict__ mean,
                                float* __restrict__ var, int M) {
  int c = threadIdx.x;
  float s = 0.f, ss = 0.f;
  for (int g = 0; g < M; ++g) { float v = t[(size_t)g * Hh + c]; s += v; ss += v * v; }
  float m = s / (float)M;
  mean[c] = m;
  var[c]  = ss / (float)M - m * m;
}

__global__ void bn_apply_relu_kernel(float* __restrict__ t, const float* __restrict__ mean,
                                     const float* __restrict__ var, const float* __restrict__ g,
                                     const float* __restrict__ b, int M) {
  int idx = blockIdx.x * blockDim.x + threadIdx.x;
  if (idx >= M * Hh) return;
  int c = idx & 63;
  float v = (t[idx] - mean[c]) * rsqrtf(var[c] + EPSv) * g[c] + b[c];
  t[idx] = fmaxf(v, 0.0f);
}

__global__ void final_out_kernel(const float* __restrict__ t, const float* __restrict__ w,
                                 const float* __restrict__ b, float* __restrict__ out, int M) {
  int g = blockIdx.x * blockDim.x + threadIdx.x;
  if (g >= M) return;
  float s = 0.f;
  for (int k = 0; k < Hh; ++k) s += t[(size_t)g * Hh + k] * w[k];
  s += b[0];
  out[g]      = s;   // first copy of tuple output
  out[Gg + g] = s;   // second copy of tuple output
}

// energy = 0.5 * sum_edges ||x[src]-x[dst]||^2 ; wave-level reduction, 1 atomic/wave
__global__ void energy_kernel(const float* __restrict__ x, const int* __restrict__ src,
                              const int* __restrict__ dst, float* __restrict__ out, int E) {
  int e = blockIdx.x * blockDim.x + threadIdx.x;
  float s = 0.f;
  if (e < E) {
    const float4* a = (const float4*)(x + (size_t)src[e] * Hh);
    const float4* b = (const float4*)(x + (size_t)dst[e] * Hh);
    for (int i = 0; i < 16; ++i) {
      float4 av = a[i], bv = b[i];
      float dx = av.x - bv.x, dy = av.y - bv.y, dz = av.z - bv.z, dw = av.w - bv.w;
      s += dx * dx + dy * dy + dz * dz + dw * dw;
    }
  }
  for (int off = 16; off > 0; off >>= 1) s += __shfl_down(s, off);
  if ((threadIdx.x & 31) == 0) atomicAdd(out, 0.5f * s);
}

// ---------------------------------------------------------------------------
static inline int cdiv(long a, long b) { return (int)((a + b - 1) / b); }

extern "C" void kernel_launch(void* const* d_in, const int* in_sizes, int n_in,
                              void* d_out, int out_size, void* d_ws, size_t ws_size,
                              hipStream_t stream) {
  const float* X     = (const float*)d_in[0];
  const int*   ei    = (const int*)  d_in[1];
  const int*   batch = (const int*)  d_in[2];
  const float* preW  = (const float*)d_in[3];
  const float* preb  = (const float*)d_in[4];
  const float* convW = (const float*)d_in[5];
  const float* convb = (const float*)d_in[6];
  const float* ggW   = (const float*)d_in[7];
  const float* ggb   = (const float*)d_in[8];
  const float* lin1W = (const float*)d_in[9];
  const float* lin1b = (const float*)d_in[10];
  const float* lin2W = (const float*)d_in[11];
  const float* lin2b = (const float*)d_in[12];
  const float* lin3W = (const float*)d_in[13];
  const float* lin3b = (const float*)d_in[14];
  const float* lin4W = (const float*)d_in[15];
  const float* lin4b = (const float*)d_in[16];
  const float* bng   = (const float*)d_in[17];
  const float* bnb   = (const float*)d_in[18];
  const int* src = ei;
  const int* dst = ei + Ee;
  float* out = (float*)d_out;

  // workspace carve-up (floats)
  float* ws   = (float*)d_ws;
  float* x    = ws;                 // N*H  current features
  float* h    = x    + Nn * Hh;     // N*H  x @ W
  float* acc  = h    + Nn * Hh;     // N*H  aggregation
  float* xn   = acc  + Nn * Hh;     // N*H  relu(conv) result
  float* ssum = xn   + Nn * Hh;     // N*H  gate diff sums (-> tau)
  float* deg  = ssum + Nn * Hh;     // N
  float* dinv = deg  + Nn;          // N
  float* cnt  = dinv + Nn;          // N    out-degree
  float* pooled = cnt + Nn;         // G*H
  float* t1   = pooled + Gg * Hh;   // G*H
  float* t2   = t1   + Gg * Hh;     // G*H
  float* mean = t2   + Gg * Hh;     // H
  float* var  = mean + Hh;          // H
  (void)in_sizes; (void)n_in; (void)out_size; (void)ws_size;

  const int TB = 256;
  const int nh = Nn * Hh;                 // 3.2M
  const int e16 = Ee * 16;                // 12.8M
  const int gemmBlocksN = cdiv(Nn, 16 * GEMM_WAVES);
  const int gemmBlocksG = cdiv(Gg, 16 * GEMM_WAVES);

  // ---- graph-static precompute ----
  fill_kernel<<<cdiv(Nn, TB), TB, 0, stream>>>(deg, 1.0f, Nn);     // self-loop
  count_kernel<<<cdiv(Ee, TB), TB, 0, stream>>>(dst, deg, Ee);
  fill_kernel<<<cdiv(Nn, TB), TB, 0, stream>>>(cnt, 0.0f, Nn);
  count_kernel<<<cdiv(Ee, TB), TB, 0, stream>>>(src, cnt, Ee);
  rsqrt_kernel<<<cdiv(Nn, TB), TB, 0, stream>>>(deg, dinv, Nn);

  // ---- prelinear: x = relu(X @ preW + preb) ----
  gemm64_wmma<<<gemmBlocksN, 32 * GEMM_WAVES, 0, stream>>>(X, preW, preb, x, Nn, DINk, FUSE_BIAS_RELU);

  // ---- G2-GCN layers ----
  for (int l = 0; l < Ll; ++l) {
    const float* cW = convW + (size_t)l * Hh * Hh;
    const float* cb = convb + (size_t)l * Hh;
    const float* gW = ggW   + (size_t)l * Hh * Hh;
    const float* gb = ggb   + (size_t)l * Hh;

    // conv branch: xn = relu(GCN(x))
    gemm64_wmma<<<gemmBlocksN, 32 * GEMM_WAVES, 0, stream>>>(x, cW, nullptr, h, Nn, Hh, FUSE_NONE);
    init_acc_kernel<<<cdiv(nh, TB), TB, 0, stream>>>(acc, h, dinv, cb, Nn);
    edge_agg_kernel<<<cdiv(e16, TB), TB, 0, stream>>>(h, dinv, src, dst, acc, Ee);
    relu_kernel<<<cdiv(nh, TB), TB, 0, stream>>>(acc, xn, nh);

    // gate branch: Xc = relu(GCN(x)) with ggW
    gemm64_wmma<<<gemmBlocksN, 32 * GEMM_WAVES, 0, stream>>>(x, gW, nullptr, h, Nn, Hh, FUSE_NONE);
    init_acc_kernel<<<cdiv(nh, TB), TB, 0, stream>>>(acc, h, dinv, gb, Nn);
    edge_agg_kernel<<<cdiv(e16, TB), TB, 0, stream>>>(h, dinv, src, dst, acc, Ee);
    relu_kernel<<<cdiv(nh, TB), TB, 0, stream>>>(acc, acc, nh);   // acc := Xc

    fill_kernel<<<cdiv(nh, TB), TB, 0, stream>>>(ssum, 0.0f, nh);
    edge_diff_kernel<<<cdiv(e16, TB), TB, 0, stream>>>(acc, src, dst, ssum, Ee);
    combine_kernel<<<cdiv(nh, TB), TB, 0, stream>>>(x, xn, ssum, cnt, Nn);
  }

  // ---- readout ----
  fill_kernel<<<cdiv(Gg * Hh, TB), TB, 0, stream>>>(pooled, 0.0f, Gg * Hh);
  pool_kernel<<<cdiv(Nn * 16, TB), TB, 0, stream>>>(x, batch, pooled, Nn);

  gemm64_wmma<<<gemmBlocksG, 32 * GEMM_WAVES, 0, stream>>>(pooled, lin1W, lin1b, t1, Gg, Hh, FUSE_BIAS);
  bn_stats_kernel<<<1, Hh, 0, stream>>>(t1, mean, var, Gg);
  bn_apply_relu_kernel<<<cdiv(Gg * Hh, TB), TB, 0, stream>>>(t1, mean, var, bng, bnb, Gg);

  gemm64_wmma<<<gemmBlocksG, 32 * GEMM_WAVES, 0, stream>>>(t1, lin2W, lin2b, t2, Gg, Hh, FUSE_BIAS);
  bn_stats_kernel<<<1, Hh, 0, stream>>>(t2, mean, var, Gg);
  bn_apply_relu_kernel<<<cdiv(Gg * Hh, TB), TB, 0, stream>>>(t2, mean, var, bng, bnb, Gg);

  gemm64_wmma<<<gemmBlocksG, 32 * GEMM_WAVES, 0, stream>>>(t2, lin3W, lin3b, t1, Gg, Hh, FUSE_BIAS);
  bn_stats_kernel<<<1, Hh, 0, stream>>>(t1, mean, var, Gg);
  bn_apply_relu_kernel<<<cdiv(Gg * Hh, TB), TB, 0, stream>>>(t1, mean, var, bng, bnb, Gg);

  final_out_kernel<<<cdiv(Gg, TB), TB, 0, stream>>>(t1, lin4W, lin4b, out, Gg);

  // ---- dirichlet energy ----
  fill_kernel<<<1, 1, 0, stream>>>(out + 2 * Gg, 0.0f, 1);
  energy_kernel<<<cdiv(Ee, TB), TB, 0, stream>>>(x, src, dst, out + 2 * Gg, Ee);
}